// CPhaseLayer_3418793967681
// MI455X (gfx1250) — compile-verified
//
#include <hip/hip_runtime.h>

// CPhase diagonal layer: out[b,c,i] = x[b,c,i] * diag[i], diag[i] in {+1,-1}.
// diag[i] = -1 iff an odd number of CPHASE qubit-pairs have both bits of i set.
// Pair q (q = parity, parity+2, ... while q+1 < n) occupies bits
// (n-1-q, n-2-q). Both-bits-set at a pair with low bit k is bit k of
// t = i & (i >> 1). The set of low-bit positions {n-2-q} is an alternating
// sequence stepping by 2 down to bit 0 or 1, so the mask is a truncated
// 0x5555/0xAAAA pattern (branchless, no loop):
//     top  = n - 1 - parity
//     base = ((n - parity) & 1) ? 0xAAAAAAAA : 0x55555555
//     mask = top > 0 ? base & ((1 << top) - 1) : 0
//     sign(i) = parity of popcount(i & (i>>1) & mask)
// Applying the sign is one XOR on the float's sign bit — no FP math at all.
//
// Memory-bound: ~1.07 GB moved -> ~46 us floor at 23.3 TB/s. b128 vector
// loads/stores with non-temporal hints (512 MB each way >> 192 MB L2,
// zero reuse).

typedef unsigned int v4u __attribute__((ext_vector_type(4)));

__global__ __launch_bounds__(256) void cphase_diag_b128_kernel(
    const v4u* __restrict__ x,
    v4u* __restrict__ out,
    const int* __restrict__ p_num_qubits,
    const int* __restrict__ p_parity,
    unsigned n4)  // number of 4-dword elements (total floats / 4)
{
    unsigned v = blockIdx.x * blockDim.x + threadIdx.x;
    if (v >= n4) return;

    // Uniform scalars (1-element device arrays) -> scalar loads.
    const int n      = p_num_qubits[0];
    const int parity = p_parity[0];

    // Branchless pair-mask (uniform SALU, ~6 ops).
    const int      top  = n - 1 - parity;
    const unsigned base = ((unsigned)(n - parity) & 1u) ? 0xAAAAAAAAu
                                                        : 0x55555555u;
    const unsigned trunc = (top > 0) ? ((1u << top) - 1u) : 0u;
    const unsigned mask  = base & trunc;

    const unsigned dmask = (1u << n) - 1u;  // n <= 20 here

    // Index within the 2^n state vector for the first of our 4 floats.
    // Flat float index = ((b*2 + c) * 2^n + i); 2^n is a multiple of 4, so
    // masking the low n bits of 4*v yields i, and i..i+3 share one (b,c) slice.
    const unsigned i0 = (v << 2) & dmask;

    const v4u d = __builtin_nontemporal_load(&x[v]);

    const unsigned a0 = i0;
    const unsigned a1 = i0 + 1u;
    const unsigned a2 = i0 + 2u;
    const unsigned a3 = i0 + 3u;

    const unsigned t0 = a0 & (a0 >> 1) & mask;
    const unsigned t1 = a1 & (a1 >> 1) & mask;
    const unsigned t2 = a2 & (a2 >> 1) & mask;
    const unsigned t3 = a3 & (a3 >> 1) & mask;

    v4u r;
    r.x = d.x ^ ((unsigned)__popc(t0) << 31);
    r.y = d.y ^ ((unsigned)__popc(t1) << 31);
    r.z = d.z ^ ((unsigned)__popc(t2) << 31);
    r.w = d.w ^ ((unsigned)__popc(t3) << 31);

    __builtin_nontemporal_store(r, &out[v]);
}

extern "C" void kernel_launch(void* const* d_in, const int* in_sizes, int n_in,
                              void* d_out, int out_size, void* d_ws, size_t ws_size,
                              hipStream_t stream) {
    (void)n_in; (void)d_ws; (void)ws_size; (void)out_size;

    const v4u* x      = (const v4u*)d_in[0];   // (64, 2, 2^20) f32
    const int* num_q  = (const int*)d_in[1];   // scalar int (1 elem)
    const int* parity = (const int*)d_in[2];   // scalar int (1 elem)
    v4u*       out    = (v4u*)d_out;

    const unsigned n_floats = (unsigned)in_sizes[0];  // 134,217,728
    const unsigned n4       = n_floats >> 2;          // 33,554,432 vec4s

    const int block = 256;                            // 8 wave32s
    const unsigned grid = (n4 + block - 1) / block;   // 131,072 blocks

    cphase_diag_b128_kernel<<<grid, block, 0, stream>>>(x, out, num_q, parity, n4);
}